// MembraneNet_60490319396958
// MI455X (gfx1250) — compile-verified
//
#include <hip/hip_runtime.h>
#include <hip/hip_bf16.h>

#define B_TOTAL 16384
#define H 64
#define L 4
#define NW 2                 // waves per block
#define THREADS (NW * 32)
#define SAMPLES_PER_BLOCK (NW * 16)

typedef float v2f __attribute__((ext_vector_type(2)));
typedef float v8f __attribute__((ext_vector_type(8)));

// Hardware tanh: CDNA5 V_TANH_F32 (TRANS pipe, co-executes with WMMA)
static __device__ inline float fast_tanh(float x) {
#if defined(__has_builtin) && __has_builtin(__builtin_amdgcn_tanhf)
    return __builtin_amdgcn_tanhf(x);
#else
    float r;
    asm volatile("v_tanh_f32 %0, %1" : "=v"(r) : "v"(x));
    return r;
#endif
}

// D = A(16x4 f32) * B(4x16 f32) + C(16x16 f32)
static __device__ inline v8f wmma4(v2f a, v2f b, v8f c) {
    return __builtin_amdgcn_wmma_f32_16x16x4_f32(
        /*neg_a=*/false, a, /*neg_b=*/false, b,
        /*c_mod=*/(short)0, c, /*reuse_a=*/false, /*reuse_b=*/false);
}

// A-fragment: rows M = mtile*16 + lid (all lanes), cols K = k0 + 2*half + {0,1}
static __device__ inline v2f ldA(const float* W, int ld,
                                 int mtile, int k0, int lid, int half) {
    const float* p = W + (mtile * 16 + lid) * ld + (k0 + 2 * half);
    v2f r; r.x = p[0]; r.y = p[1];
    return r;
}

// B-fragment from transposed activation tile X[64][16]: (K = k0+2*half+{0,1}, N = lid)
static __device__ inline v2f ldB(const float* X,
                                 int k0, int lid, int half) {
    int kk = k0 + 2 * half;
    v2f r; r.x = X[kk * 16 + lid]; r.y = X[(kk + 1) * 16 + lid];
    return r;
}

__global__ __launch_bounds__(THREADS)
void membrane_kernel(const float* __restrict__ xy,
                     const float* __restrict__ Kc,
                     const float* __restrict__ Kx,
                     const float* __restrict__ Ky,
                     const float* __restrict__ fin,
                     const float* __restrict__ W0,  // (64,2)
                     const float* __restrict__ b0,
                     const float* __restrict__ W1,  // (64,64)
                     const float* __restrict__ b1,
                     const float* __restrict__ W2,
                     const float* __restrict__ b2,
                     const float* __restrict__ W3,
                     const float* __restrict__ b3,
                     const float* __restrict__ Wout,
                     const float* __restrict__ boutp,
                     float* __restrict__ out) {
    __shared__ __align__(16) float sW[3][H * H];     // W1,W2,W3 row-major
    __shared__ __align__(16) float sW0[H * 4];       // W0 padded K 2->4
    __shared__ float sB[4][H];
    __shared__ float sWout[H];
    __shared__ __align__(16) float sH[NW][L][H * 16];      // tanh outputs, [neuron][sample]
    __shared__ __align__(16) float sA[NW][L][H * 16];      // backprop vectors a_k
    __shared__ __align__(16) float sQ[NW][2][2][H * 16];   // [buf][m][neuron][sample]
    __shared__ __align__(16) float sXY[NW][4 * 16];        // layer-0 input, padded

    const int tid  = threadIdx.x;
    const int lane = tid & 31;
    const int wave = tid >> 5;
    const int half = lane >> 4;      // 0: lanes 0-15, 1: lanes 16-31
    const int lid  = lane & 15;      // sample-in-wave / A-row index

    // ---- cooperative weight load ----
    for (int idx = tid; idx < H * H; idx += THREADS) {
        sW[0][idx] = W1[idx];
        sW[1][idx] = W2[idx];
        sW[2][idx] = W3[idx];
    }
    for (int idx = tid; idx < H * 4; idx += THREADS) {
        int j = idx >> 2, c = idx & 3;
        sW0[idx] = (c < 2) ? W0[j * 2 + c] : 0.0f;
    }
    for (int idx = tid; idx < H; idx += THREADS) {
        sB[0][idx] = b0[idx];
        sB[1][idx] = b1[idx];
        sB[2][idx] = b2[idx];
        sB[3][idx] = b3[idx];
        sWout[idx] = Wout[idx];
    }

    // ---- per-wave layer-0 input tile (transposed, padded) ----
    const int gbase = blockIdx.x * SAMPLES_PER_BLOCK + wave * 16;
    if (half == 0) {
        int g = gbase + lid;
        sXY[wave][0 * 16 + lid] = xy[2 * g + 0];
        sXY[wave][1 * 16 + lid] = xy[2 * g + 1];
    } else {
        sXY[wave][2 * 16 + lid] = 0.0f;
        sXY[wave][3 * 16 + lid] = 0.0f;
    }
    __syncthreads();

    // =========== forward pass: h_l, stored transposed ===========
    {
        const float* src = sXY[wave];
        for (int l = 0; l < L; ++l) {
            const float* Wmat = (l == 0) ? sW0 : sW[l - 1];
            const int ld   = (l == 0) ? 4 : H;
            const int Kdim = (l == 0) ? 4 : H;
            v8f acc[4];
            #pragma unroll
            for (int mt = 0; mt < 4; ++mt)
                #pragma unroll
                for (int r = 0; r < 8; ++r)
                    acc[mt][r] = sB[l][mt * 16 + half * 8 + r];
            for (int k0 = 0; k0 < Kdim; k0 += 4) {
                v2f bf = ldB(src, k0, lid, half);
                #pragma unroll
                for (int mt = 0; mt < 4; ++mt)
                    acc[mt] = wmma4(ldA(Wmat, ld, mt, k0, lid, half), bf, acc[mt]);
            }
            #pragma unroll
            for (int mt = 0; mt < 4; ++mt)
                #pragma unroll
                for (int r = 0; r < 8; ++r) {
                    float h = fast_tanh(acc[mt][r]);
                    sH[wave][l][(mt * 16 + half * 8 + r) * 16 + lid] = h;
                }
            src = sH[wave][l];
        }
    }

    // =========== backprop vectors a_k = (a_{k+1} . d_{k+1}) @ W_{k+1} ===========
    for (int idx = lane; idx < H * 16; idx += 32)
        sA[wave][L - 1][idx] = sWout[idx >> 4];
    for (int k = L - 2; k >= 0; --k) {
        const float* Wn = sW[k];                 // == W_{k+1}
        const float* ap = sA[wave][k + 1];
        const float* hp = sH[wave][k + 1];
        v8f acc[4] = {};
        for (int k0 = 0; k0 < H; k0 += 4) {
            const int i0 = k0 + 2 * half;
            v2f bf;
            {
                float av = ap[i0 * 16 + lid], hv = hp[i0 * 16 + lid];
                bf.x = av * (1.0f - hv * hv);
            }
            {
                float av = ap[(i0 + 1) * 16 + lid], hv = hp[(i0 + 1) * 16 + lid];
                bf.y = av * (1.0f - hv * hv);
            }
            #pragma unroll
            for (int mt = 0; mt < 4; ++mt) {
                v2f af;                           // A = Wn^T : (M=j, K=i) = Wn[i][j]
                af.x = Wn[i0 * H + mt * 16 + lid];
                af.y = Wn[(i0 + 1) * H + mt * 16 + lid];
                acc[mt] = wmma4(af, bf, acc[mt]);
            }
        }
        #pragma unroll
        for (int mt = 0; mt < 4; ++mt)
            #pragma unroll
            for (int r = 0; r < 8; ++r)
                sA[wave][k][(mt * 16 + half * 8 + r) * 16 + lid] = acc[mt][r];
    }

    // =========== Q forward + Hessian/gradient accumulation ===========
    for (int idx = lane; idx < H * 16; idx += 32) {
        int j = idx >> 4;
        sQ[wave][0][0][idx] = sW0[j * 4 + 0];
        sQ[wave][0][1][idx] = sW0[j * 4 + 1];
    }
    float uxx = 0.f, uyy = 0.f, dux = 0.f, duy = 0.f, uval = 0.f;
    int cur = 0;
    const int jbase = half * 32;
    for (int k = 0; k < L; ++k) {
        const float* ap = sA[wave][k];
        const float* hp = sH[wave][k];
        const float* q0 = sQ[wave][cur][0];
        const float* q1 = sQ[wave][cur][1];
        for (int jj = 0; jj < 32; ++jj) {
            const int id = (jbase + jj) * 16 + lid;
            float a = ap[id], h = hp[id];
            float d = 1.0f - h * h;
            float t = a * (-2.0f * h * d);
            float x0 = q0[id], x1 = q1[id];
            uxx += t * x0 * x0;
            uyy += t * x1 * x1;
            if (k == 0) { dux += a * d * x0; duy += a * d * x1; }
        }
        if (k < L - 1) {
            const float* Wn = sW[k];             // == W_{k+1}
            #pragma unroll
            for (int m = 0; m < 2; ++m) {
                const float* qs = sQ[wave][cur][m];
                v8f acc[4] = {};
                for (int k0 = 0; k0 < H; k0 += 4) {
                    const int i0 = k0 + 2 * half;
                    v2f bf;
                    {
                        float hv = hp[i0 * 16 + lid];
                        bf.x = (1.0f - hv * hv) * qs[i0 * 16 + lid];
                    }
                    {
                        float hv = hp[(i0 + 1) * 16 + lid];
                        bf.y = (1.0f - hv * hv) * qs[(i0 + 1) * 16 + lid];
                    }
                    #pragma unroll
                    for (int mt = 0; mt < 4; ++mt)
                        acc[mt] = wmma4(ldA(Wn, H, mt, k0, lid, half), bf, acc[mt]);
                }
                #pragma unroll
                for (int mt = 0; mt < 4; ++mt)
                    #pragma unroll
                    for (int r = 0; r < 8; ++r)
                        sQ[wave][cur ^ 1][m][(mt * 16 + half * 8 + r) * 16 + lid] = acc[mt][r];
            }
            cur ^= 1;
        }
    }

    // output head: u = h3 . Wout + bout
    {
        const float* h3 = sH[wave][L - 1];
        for (int jj = 0; jj < 32; ++jj) {
            int j = jbase + jj;
            uval += sWout[j] * h3[j * 16 + lid];
        }
    }

    // combine the two lane-halves (wave32)
    uxx += __shfl_xor(uxx, 16);
    uyy += __shfl_xor(uyy, 16);
    dux += __shfl_xor(dux, 16);
    duy += __shfl_xor(duy, 16);
    uval += __shfl_xor(uval, 16);

    if (half == 0) {
        int g = gbase + lid;
        float u = uval + boutp[0];
        float pde = Kc[g] * (uxx + uyy) + Kx[g] * dux + Ky[g] * duy + fin[g];
        out[g] = u;
        out[B_TOTAL + g] = pde;
    }
}

extern "C" void kernel_launch(void* const* d_in, const int* in_sizes, int n_in,
                              void* d_out, int out_size, void* d_ws, size_t ws_size,
                              hipStream_t stream) {
    (void)in_sizes; (void)n_in; (void)d_ws; (void)ws_size; (void)out_size;
    const float* xy   = (const float*)d_in[0];
    const float* Kc   = (const float*)d_in[1];
    const float* Kx   = (const float*)d_in[2];
    const float* Ky   = (const float*)d_in[3];
    const float* fin  = (const float*)d_in[4];
    const float* W0   = (const float*)d_in[5];
    const float* b0   = (const float*)d_in[6];
    const float* W1   = (const float*)d_in[7];
    const float* b1   = (const float*)d_in[8];
    const float* W2   = (const float*)d_in[9];
    const float* b2   = (const float*)d_in[10];
    const float* W3   = (const float*)d_in[11];
    const float* b3   = (const float*)d_in[12];
    const float* Wout = (const float*)d_in[13];
    const float* bout = (const float*)d_in[14];
    float* out = (float*)d_out;

    dim3 grid(B_TOTAL / SAMPLES_PER_BLOCK);   // 512 blocks
    dim3 block(THREADS);                      // 2 waves
    hipLaunchKernelGGL(membrane_kernel, grid, block, 0, stream,
                       xy, Kc, Kx, Ky, fin, W0, b0, W1, b1, W2, b2, W3, b3,
                       Wout, bout, out);
}